// MixtureOfLogistics_62861141344576
// MI455X (gfx1250) — compile-verified
//
#include <hip/hip_runtime.h>
#include <hip/hip_bf16.h>

// Problem constants (match reference exactly)
#define K_CENTERS 64
#define BATCH     128
#define ESIZE     3072            // 3*32*32
#define EB        256             // elements (pixels) per block
#define NEB       (ESIZE / EB)    // 12 element chunks
#define BT        8               // batches per block
#define NBT       (BATCH / BT)    // 16 batch tiles

#define XT_SCALE   0.5019607843137255f   // (x*64 + 128 - 128)/255*2 = x*128/255
#define BIN_HALF   0.00392156862745098f  // 1/255
#define LOG_127_5  4.848116235f          // ln(127.5)

typedef __attribute__((ext_vector_type(2))) float v2f;
typedef __attribute__((ext_vector_type(8))) float v8f;

// ---- CDNA5 async global->LDS staging (guarded: falls back to direct loads) --
#if defined(__HIP_DEVICE_COMPILE__) && \
    __has_builtin(__builtin_amdgcn_global_load_async_to_lds_b32)
#define HAVE_ASYNC_LDS 1
#else
#define HAVE_ASYNC_LDS 0
#endif

#if HAVE_ASYNC_LDS
typedef __attribute__((address_space(1))) int gas_int;   // global (AS1)
typedef __attribute__((address_space(3))) int las_int;   // LDS (AS3)

__device__ __forceinline__ void stage_async_b32(const float* gptr, float* lptr) {
    __builtin_amdgcn_global_load_async_to_lds_b32(
        (gas_int*)gptr, (las_int*)lptr, /*offset=*/0, /*cpol=*/0);
}
__device__ __forceinline__ void wait_async0() {
#if __has_builtin(__builtin_amdgcn_s_wait_asynccnt)
    __builtin_amdgcn_s_wait_asynccnt(0);
#else
    asm volatile("s_wait_asynccnt 0" ::: "memory");
#endif
    asm volatile("" ::: "memory");   // compiler barrier: LDS reads stay after wait
}
#endif

// Fast hardware transcendentals: v_exp_f32 is 2^x, v_log_f32 is log2(x).
__device__ __forceinline__ float fexp(float x) {
    return __builtin_amdgcn_exp2f(x * 1.4426950408889634f);
}
__device__ __forceinline__ float flog(float x) {
    return __builtin_amdgcn_logf(x) * 0.6931471805599453f;
}

// Stable softplus + sigmoid sharing one v_exp:
//   e = exp(-|t|) in (0,1]
//   softplus(t) = max(t,0) + log1p(e)
//   sigmoid(t)  = t>=0 ? 1/(1+e) : e/(1+e)
__device__ __forceinline__ void softplus_sigmoid(float t, float& sp, float& sig) {
    float a = __builtin_fabsf(t);
    float e = fexp(-a);
    float l = flog(1.0f + e);
    sp = __builtin_fmaxf(t, 0.0f) + l;
    float r = __builtin_amdgcn_rcpf(1.0f + e);
    sig = (t >= 0.0f) ? r : e * r;
}

__device__ __forceinline__ float softplus_only(float t) {
    float a = __builtin_fabsf(t);
    return __builtin_fmaxf(t, 0.0f) + flog(1.0f + fexp(-a));
}

// ---------------------------------------------------------------------------
// Kernel 1: per-(element-chunk, batch-tile) partial sums over all centers.
// mu/log_s k-slices are double-buffered into LDS with async global->LDS copies
// (each lane stages its own element, so ASYNCcnt alone synchronizes — no
// block barrier in the k-loop). Transcendental-bound; 8 independent batch
// lanes per thread give the TRANS pipe ILP to co-execute with VALU.
// ---------------------------------------------------------------------------
__global__ __launch_bounds__(256) void mol_partial_kernel(
    const float* __restrict__ x, const float* __restrict__ mask,
    const float* __restrict__ log_s, const float* __restrict__ mu,
    float* __restrict__ part)
{
    const int tid = threadIdx.x;
    const int ec  = blockIdx.x;          // element chunk 0..11
    const int bb  = blockIdx.y * BT;     // first batch of tile
    const int e   = ec * EB + tid;       // this thread's pixel

    float xv[BT], mv[BT], acc[BT];
#pragma unroll
    for (int j = 0; j < BT; ++j) {
        xv[j]  = x[(bb + j) * ESIZE + e] * XT_SCALE;   // xt in [-1,1] space
        mv[j]  = mask[(bb + j) * ESIZE + e];
        acc[j] = 0.0f;
    }

#if HAVE_ASYNC_LDS
    __shared__ float s_mu[2][EB];
    __shared__ float s_ls[2][EB];
    // prime buffer 0 with the k=0 slices
    stage_async_b32(&mu[e],    &s_mu[0][tid]);
    stage_async_b32(&log_s[e], &s_ls[0][tid]);
    wait_async0();
#endif

    for (int k = 0; k < K_CENTERS; ++k) {
#if HAVE_ASYNC_LDS
        const int buf = k & 1;
        if (k + 1 < K_CENTERS) {          // kick off k+1 while computing k
            stage_async_b32(&mu[(k + 1) * ESIZE + e],    &s_mu[buf ^ 1][tid]);
            stage_async_b32(&log_s[(k + 1) * ESIZE + e], &s_ls[buf ^ 1][tid]);
        }
        const float mu_ke = s_mu[buf][tid];
        const float ls_ke = s_ls[buf][tid];
#else
        const float mu_ke = mu[k * ESIZE + e];
        const float ls_ke = log_s[k * ESIZE + e];
        if (k + 1 < K_CENTERS) {
            // gfx1250 global_prefetch_b8 for the next center's slices
            __builtin_prefetch(&mu[(k + 1) * ESIZE + e], 0, 0);
            __builtin_prefetch(&log_s[(k + 1) * ESIZE + e], 0, 0);
        }
#endif
        const float inv = fexp(-ls_ke);   // exp(-log_s): once per (k,e)

#pragma unroll
        for (int j = 0; j < BT; ++j) {
            const float xt  = xv[j];
            const float cx  = xt - mu_ke;
            const float p   = inv * (cx + BIN_HALF);
            const float mn  = inv * (cx - BIN_HALF);
            const float mid = inv * cx;

            float sp_p, sig_p, sp_m, sig_m;
            softplus_sigmoid(p,  sp_p, sig_p);
            softplus_sigmoid(mn, sp_m, sig_m);
            const float sp_mid = softplus_only(mid);

            const float cdf_delta       = sig_p - sig_m;
            const float log_cdf_plus    = p - sp_p;            // log sigmoid(p)
            const float log_om_cdf_min  = -sp_m;               // log(1-sigmoid(m))
            const float log_pdf_mid     = mid - cx - 2.0f * sp_mid;

            const float lg = flog(__builtin_fmaxf(cdf_delta, 1e-12f));
            float lp = (cdf_delta > 1e-5f) ? lg : (log_pdf_mid - LOG_127_5);
            lp = (xt >  0.999f) ? log_om_cdf_min : lp;
            lp = (xt < -0.999f) ? log_cdf_plus   : lp;

            acc[j] = __builtin_fmaf(lp, mv[j], acc[j]);
        }

#if HAVE_ASYNC_LDS
        wait_async0();   // k+1 slices resident before next iteration reads them
#endif
    }

    // Reduce 256 threads -> 8 batch partials. Wave w owns batch-lane w.
    __shared__ float red[BT * EB];   // 8 KB
#pragma unroll
    for (int j = 0; j < BT; ++j) red[j * EB + tid] = acc[j];
    __syncthreads();

    const int w    = tid >> 5;   // wave index 0..7 == batch lane
    const int lane = tid & 31;
    float s = 0.0f;
#pragma unroll
    for (int i = 0; i < EB / 32; ++i) s += red[w * EB + lane + i * 32];
#pragma unroll
    for (int off = 16; off > 0; off >>= 1) s += __shfl_xor(s, off, 32);

    if (lane == 0) part[ec * BATCH + bb + w] = s;
}

// ---------------------------------------------------------------------------
// Kernel 2: deterministic WMMA column-sum of the [NEB=12][BATCH=128] partial
// matrix, + log-softmax-weight constant.
//   D = ones[16x4] x P[4x16] + C  ->  every row of D = column sums of P.
// Wave w (of 8) handles batches [16w, 16w+16); 3 accumulating WMMAs cover the
// 12 rows. Fixed accumulation order -> deterministic (unlike float atomics).
// ---------------------------------------------------------------------------
__global__ __launch_bounds__(256) void mol_reduce_wmma_kernel(
    const float* __restrict__ part, const float* __restrict__ log_weight,
    float* __restrict__ out)
{
    const int tid  = threadIdx.x;
    const int w    = tid >> 5;          // column group (16 batches each)
    const int lane = tid & 31;
    const int col  = w * 16 + (lane & 15);
    const int half = lane >> 4;         // which pair of K-rows this half-wave holds

    v2f a; a.x = 1.0f; a.y = 1.0f;      // all-ones A (16x4)
    v8f c = {};                          // accumulator C/D (16x16 f32)

#pragma unroll
    for (int kk = 0; kk < NEB / 4; ++kk) {      // 3 chunks of 4 rows = 12 rows
        const int r0 = kk * 4 + half * 2;
        v2f b;
        b.x = part[(r0 + 0) * BATCH + col];
        b.y = part[(r0 + 1) * BATCH + col];
        // v_wmma_f32_16x16x4_f32 : D = A*B + C
        c = __builtin_amdgcn_wmma_f32_16x16x4_f32(
                /*neg_a=*/false, a, /*neg_b=*/false, b,
                /*c_mod=*/(short)0, c, /*reuse_a=*/false, /*reuse_b=*/false);
    }

    // Constant: sum_k log_softmax(log_weight)[k] = sum(lw) - K * logsumexp(lw)
    float mx = -3.4028235e38f;
    for (int i = 0; i < K_CENTERS; ++i) mx = __builtin_fmaxf(mx, log_weight[i]);
    float tot = 0.0f, se = 0.0f;
    for (int i = 0; i < K_CENTERS; ++i) {
        const float v = log_weight[i];
        tot += v;
        se  += fexp(v - mx);
    }
    const float lconst = tot - (float)K_CENTERS * (mx + flog(se));

    // D-matrix layout: VGPR0, lanes 0..15 = row M=0, N=lane -> column sums.
    if (lane < 16) out[w * 16 + lane] = c[0] + lconst;
}

// ---------------------------------------------------------------------------
extern "C" void kernel_launch(void* const* d_in, const int* in_sizes, int n_in,
                              void* d_out, int out_size, void* d_ws, size_t ws_size,
                              hipStream_t stream) {
    (void)in_sizes; (void)n_in; (void)out_size; (void)ws_size;
    const float* x     = (const float*)d_in[0];   // [128,3,32,32]
    const float* mask  = (const float*)d_in[1];   // [128,3,32,32]
    const float* lw    = (const float*)d_in[2];   // [64]
    const float* log_s = (const float*)d_in[3];   // [64,3,32,32]
    const float* mu    = (const float*)d_in[4];   // [64,3,32,32]
    float* out  = (float*)d_out;                  // [128]
    float* part = (float*)d_ws;                   // [12][128] partials (6 KB)

    dim3 grid(NEB, NBT);                          // 12 x 16 = 192 blocks
    mol_partial_kernel<<<grid, 256, 0, stream>>>(x, mask, log_s, mu, part);
    mol_reduce_wmma_kernel<<<1, 256, 0, stream>>>(part, lw, out);
}